// GCN_new_52115133170062
// MI455X (gfx1250) — compile-verified
//
#include <hip/hip_runtime.h>
#include <hip/hip_bf16.h>

// ---------------------------------------------------------------------------
// GCN forward (4 layers) for MI455X / gfx1250.
// Dominant cost: edge gather (L2-resident h rows) + fp32 atomic scatter.
// GEMMs: V_WMMA_F32_16X16X4_F32 (fp32 WMMA, wave32), branch-free inner loop
// via compile-time NCOLS and zero-padded W3 (40 -> 48 cols).
// ---------------------------------------------------------------------------

typedef __attribute__((ext_vector_type(2))) float v2f;
typedef __attribute__((ext_vector_type(8))) float v8f;

#define D_FEAT 128
#define C_OUT  40
#define C_PAD  48   // 3 full 16-col WMMA tiles

// ---------------------------------------------------------------------------
// Degree count: deg[dst[e]] += 1 for every edge.
// ---------------------------------------------------------------------------
__global__ void deg_kernel(const int* __restrict__ dst, int E, int* __restrict__ deg) {
    for (int e = blockIdx.x * blockDim.x + threadIdx.x; e < E;
         e += gridDim.x * blockDim.x) {
        atomicAdd(&deg[dst[e]], 1);
    }
}

// dinv  = deg>0 ? rsqrt(deg)   : 0        (hidden layers, no self loops)
// dinvl = rsqrt(deg+1)                    (last layer, with self loop)
__global__ void dinv_kernel(const int* __restrict__ deg, int n,
                            float* __restrict__ dinv, float* __restrict__ dinvl) {
    for (int i = blockIdx.x * blockDim.x + threadIdx.x; i < n;
         i += gridDim.x * blockDim.x) {
        int d = deg[i];
        dinv[i]  = (d > 0) ? rsqrtf((float)d) : 0.0f;
        dinvl[i] = rsqrtf((float)(d + 1));
    }
}

// Zero-pad W3 (128 x 40) into Wpad (128 x 48).
__global__ void padW3_kernel(const float* __restrict__ W3, float* __restrict__ Wpad) {
    int idx = blockIdx.x * blockDim.x + threadIdx.x;   // 128*48 = 6144 threads
    if (idx >= D_FEAT * C_PAD) return;
    int row = idx / C_PAD;
    int col = idx % C_PAD;
    Wpad[idx] = (col < C_OUT) ? W3[row * C_OUT + col] : 0.0f;
}

// ---------------------------------------------------------------------------
// GEMM: Dm[nrows x NCOLS] = A[nrows x 128] * B[128 x NCOLS]
// One wave (32 lanes) computes one 16x16 output tile via v_wmma_f32_16x16x4_f32.
// nrows multiple of 16 (N = 100000 = 6250*16); NCOLS multiple of 16, so the
// inner loop is fully unconditional: pointer + immediate-offset loads only.
//
// A 16x4 f32 layout (ISA 7.12.2): lanes 0-15 hold M=lane, VGPR0=K0,VGPR1=K1;
// lanes 16-31 hold M=lane-16, VGPR0=K2,VGPR1=K3.
// B 4x16: K striped across VGPRs (second half-wave takes K+2), cols across
// lanes. C/D 16x16: VGPR r -> M=r (lanes 0-15) / M=r+8 (lanes 16-31).
// ---------------------------------------------------------------------------
template <int NCOLS>
__global__ void gemm_wmma_f32(const float* __restrict__ A,
                              const float* __restrict__ B,
                              float* __restrict__ Dm) {
    const int lane = threadIdx.x;          // 0..31
    const int row0 = blockIdx.x * 16;
    const int col0 = blockIdx.y * 16;
    const int half = lane >> 4;            // 0 or 1
    const int l    = lane & 15;
    const int koff = half * 2;

    const float* ap = A + (size_t)(row0 + l) * D_FEAT + koff;
    const float* bp = B + (size_t)koff * NCOLS + (col0 + l);

    v8f c = {};
    #pragma unroll 8
    for (int k = 0; k < D_FEAT; k += 4) {
        v2f a;
        a.x = ap[0];
        a.y = ap[1];
        v2f b;
        b.x = bp[0];
        b.y = bp[NCOLS];
        // 8 args: (neg_a, A, neg_b, B, c_mod, C, reuse_a, reuse_b)
        c = __builtin_amdgcn_wmma_f32_16x16x4_f32(
                false, a, false, b, (short)0, c, false, false);
        ap += 4;
        bp += 4 * NCOLS;
    }

    float* dp = Dm + (size_t)(row0 + half * 8) * NCOLS + (col0 + l);
    #pragma unroll
    for (int r = 0; r < 8; ++r) {
        dp[(size_t)r * NCOLS] = c[r];
    }
}

// ---------------------------------------------------------------------------
// Edge scatter, 128 features: one wave per edge, float4 per lane.
// agg[dst] += h[src] * (dinv[src]*dinv[dst])
// ---------------------------------------------------------------------------
__global__ void scatter128_kernel(const float* __restrict__ h,
                                  const int* __restrict__ src,
                                  const int* __restrict__ dst,
                                  const float* __restrict__ dinv,
                                  int E, float* __restrict__ agg) {
    const int wid  = (blockIdx.x * blockDim.x + threadIdx.x) >> 5;
    const int lane = threadIdx.x & 31;
    if (wid >= E) return;
    const int s = src[wid];
    const int d = dst[wid];
    const float norm = dinv[s] * dinv[d];

    const float4 v = ((const float4*)(h + (size_t)s * D_FEAT))[lane];
    float* arow = agg + (size_t)d * D_FEAT + lane * 4;
    atomicAdd(arow + 0, v.x * norm);
    atomicAdd(arow + 1, v.y * norm);
    atomicAdd(arow + 2, v.z * norm);
    atomicAdd(arow + 3, v.w * norm);
}

// ---------------------------------------------------------------------------
// Edge scatter, final layer: h has padded stride C_PAD, out has stride 40.
// ---------------------------------------------------------------------------
__global__ void scatter40_kernel(const float* __restrict__ h,
                                 const int* __restrict__ src,
                                 const int* __restrict__ dst,
                                 const float* __restrict__ dinvl,
                                 int E, float* __restrict__ out) {
    const int wid  = (blockIdx.x * blockDim.x + threadIdx.x) >> 5;
    const int lane = threadIdx.x & 31;
    if (wid >= E) return;
    const int s = src[wid];
    const int d = dst[wid];
    const float norm = dinvl[s] * dinvl[d];

    const float* hrow = h + (size_t)s * C_PAD;
    float* orow = out + (size_t)d * C_OUT;
    atomicAdd(orow + lane, hrow[lane] * norm);
    if (lane < C_OUT - 32) atomicAdd(orow + 32 + lane, hrow[32 + lane] * norm);
}

// ---------------------------------------------------------------------------
// PairNorm pass 1: per-column sums (stats[0..127]) + total sum of squares
// (stats[128]). blockDim must be 128 (thread j = column j).
// ---------------------------------------------------------------------------
__global__ void pn_reduce_kernel(const float* __restrict__ x, int n,
                                 float* __restrict__ stats) {
    const int col = threadIdx.x;
    float s = 0.0f, sq = 0.0f;
    for (int i = blockIdx.x; i < n; i += gridDim.x) {
        float v = x[(size_t)i * D_FEAT + col];
        s  += v;
        sq += v * v;
    }
    atomicAdd(&stats[col], s);
    atomicAdd(&stats[D_FEAT], sq);
}

// PairNorm pass 2 (one block of 128 threads):
//   mean[j] = colsum[j]/n  (stored over stats[0..127])
//   stats[129] = rsqrt(1e-6 + (sumsq - n*sum(mean^2))/n)
__global__ void pn_finalize_kernel(float* __restrict__ stats, int n) {
    __shared__ float sh[D_FEAT];
    const int j = threadIdx.x;
    const float invn = 1.0f / (float)n;
    float mean = stats[j] * invn;
    stats[j] = mean;
    sh[j] = mean * mean;
    __syncthreads();
    for (int st = 64; st > 0; st >>= 1) {
        if (j < st) sh[j] += sh[j + st];
        __syncthreads();
    }
    if (j == 0) {
        float sumsq = stats[D_FEAT];
        float var = (sumsq - (float)n * sh[0]) * invn;  // mean row-norm^2
        stats[D_FEAT + 1] = rsqrtf(1e-6f + var);
    }
}

// PairNorm pass 3 + ReLU (in place): x = relu((x - mean[col]) * invscale)
__global__ void pn_apply_kernel(float* __restrict__ x, size_t total,
                                const float* __restrict__ stats) {
    const float invscale = stats[D_FEAT + 1];
    for (size_t idx = blockIdx.x * (size_t)blockDim.x + threadIdx.x; idx < total;
         idx += (size_t)gridDim.x * blockDim.x) {
        const int col = (int)(idx & (D_FEAT - 1));
        float v = (x[idx] - stats[col]) * invscale;
        x[idx] = fmaxf(v, 0.0f);
    }
}

// Final-layer init: out[i][c] = b3[c] + h[i][c] * dinvl[i]^2   (self loop + bias)
// h has padded stride C_PAD; out has stride C_OUT.
__global__ void final_init_kernel(const float* __restrict__ h,
                                  const float* __restrict__ dinvl,
                                  const float* __restrict__ b3,
                                  float* __restrict__ out, int n) {
    const size_t total = (size_t)n * C_OUT;
    for (size_t idx = blockIdx.x * (size_t)blockDim.x + threadIdx.x; idx < total;
         idx += (size_t)gridDim.x * blockDim.x) {
        const int i = (int)(idx / C_OUT);
        const int c = (int)(idx % C_OUT);
        const float dl = dinvl[i];
        out[idx] = b3[c] + h[(size_t)i * C_PAD + c] * dl * dl;
    }
}

// ---------------------------------------------------------------------------
// Host-side orchestration (all on `stream`, graph-capture safe).
// ---------------------------------------------------------------------------
extern "C" void kernel_launch(void* const* d_in, const int* in_sizes, int n_in,
                              void* d_out, int out_size, void* d_ws, size_t ws_size,
                              hipStream_t stream) {
    const float* x   = (const float*)d_in[0];
    const int*   ei  = (const int*)d_in[1];
    const float* W0  = (const float*)d_in[2];
    const float* W1  = (const float*)d_in[3];
    const float* W2  = (const float*)d_in[4];
    const float* W3  = (const float*)d_in[5];
    // b0..b2 cancel exactly under PairNorm centering (and are zeros anyway)
    const float* b3  = (const float*)d_in[9];
    float* out = (float*)d_out;

    const int N = in_sizes[0] / D_FEAT;      // 100000
    const int E = in_sizes[1] / 2;           // 1600000
    const int* src = ei;
    const int* dst = ei + E;

    // ---- workspace layout (256B aligned) ----
    char* ws = (char*)d_ws;
    size_t off = 0;
    auto alloc = [&](size_t bytes) {
        void* p = ws + off;
        off += (bytes + 255) & ~(size_t)255;
        return p;
    };
    float* h    = (float*)alloc((size_t)N * D_FEAT * sizeof(float)); // reused N*C_PAD
    float* agg  = (float*)alloc((size_t)N * D_FEAT * sizeof(float));
    int*   deg  = (int*)  alloc((size_t)N * sizeof(int));
    float* dinv = (float*)alloc((size_t)N * sizeof(float));
    float* dinvl= (float*)alloc((size_t)N * sizeof(float));
    float* stats= (float*)alloc(130 * sizeof(float));
    float* Wpad = (float*)alloc((size_t)D_FEAT * C_PAD * sizeof(float));
    (void)ws_size;

    // ---- degrees / inverse sqrt degrees (recomputed every call: deterministic)
    hipMemsetAsync(deg, 0, (size_t)N * sizeof(int), stream);
    deg_kernel<<<2048, 256, 0, stream>>>(dst, E, deg);
    dinv_kernel<<<(N + 255) / 256, 256, 0, stream>>>(deg, N, dinv, dinvl);
    padW3_kernel<<<(D_FEAT * C_PAD + 255) / 256, 256, 0, stream>>>(W3, Wpad);

    const int rowTiles = N / 16;                 // 6250, exact
    const int scatterBlocks = (E * 32 + 255) / 256;
    const size_t totalHidden = (size_t)N * D_FEAT;

    const float* xin = x;
    const float* Ws[3] = {W0, W1, W2};
    for (int layer = 0; layer < 3; ++layer) {
        // h = xin @ W
        gemm_wmma_f32<D_FEAT><<<dim3(rowTiles, D_FEAT / 16), 32, 0, stream>>>(
            xin, Ws[layer], h);
        // agg = scatter-add of normalized messages
        hipMemsetAsync(agg, 0, totalHidden * sizeof(float), stream);
        scatter128_kernel<<<scatterBlocks, 256, 0, stream>>>(
            h, src, dst, dinv, E, agg);
        // PairNorm + ReLU in place on agg
        hipMemsetAsync(stats, 0, 130 * sizeof(float), stream);
        pn_reduce_kernel<<<2048, D_FEAT, 0, stream>>>(agg, N, stats);
        pn_finalize_kernel<<<1, D_FEAT, 0, stream>>>(stats, N);
        pn_apply_kernel<<<4096, 256, 0, stream>>>(agg, totalHidden, stats);
        xin = agg;
    }

    // ---- final layer: 128 -> 40 (padded to 48), self loops, bias ----
    gemm_wmma_f32<C_PAD><<<dim3(rowTiles, C_PAD / 16), 32, 0, stream>>>(xin, Wpad, h);
    final_init_kernel<<<4096, 256, 0, stream>>>(h, dinvl, b3, out, N);
    scatter40_kernel<<<scatterBlocks, 256, 0, stream>>>(h, src, dst, dinvl, E, out);
}